// PureTriadicBrain_73753178407521
// MI455X (gfx1250) — compile-verified
//
#include <hip/hip_runtime.h>
#include <math.h>

typedef __attribute__((ext_vector_type(2))) float v2f;
typedef __attribute__((ext_vector_type(8))) float v8f;

#define BB 8
#define NB 1024
#define DDIM 32
#define BND (BB*NB*DDIM)      // 262144
#define BNN (BB*NB*NB)        // 8388608

__device__ __forceinline__ float wave_sum(float v) {
  #pragma unroll
  for (int m = 16; m >= 1; m >>= 1) v += __shfl_xor(v, m, 32);
  return v;
}
__device__ __forceinline__ float wave_max(float v) {
  #pragma unroll
  for (int m = 16; m >= 1; m >>= 1) v = fmaxf(v, __shfl_xor(v, m, 32));
  return v;
}

// ---------------- piece_forward: q,k,pred = buntanh(x @ w) per node ----------------
__global__ void k_piece(const float* __restrict__ state,
                        const float* __restrict__ w1, const float* __restrict__ w2,
                        const float* __restrict__ w3,
                        float* __restrict__ pred, float* __restrict__ kbuf,
                        float* __restrict__ qbuf) {
  int wave = (blockIdx.x * blockDim.x + threadIdx.x) >> 5;
  int lane = threadIdx.x & 31;
  int l = wave / (BB * NB);
  int node = wave - l * (BB * NB);
  if (l >= 3) return;
  const float* w = (l == 0) ? w1 : (l == 1) ? w2 : w3;
  float* out = (l == 0) ? pred : (l == 1) ? kbuf : qbuf;
  float x = state[node * DDIM + lane];
  const float* wp = w + (size_t)node * DDIM * DDIM;
  float acc = 0.f;
  #pragma unroll
  for (int i = 0; i < DDIM; ++i)
    acc = fmaf(__shfl(x, i, 32), wp[i * DDIM + lane], acc);
  // buntanh(x, 0.5, sqrt(2+sqrt2))
  out[node * DDIM + lane] = 1.8477590650225735f * (tanhf(acc) + 0.5f * acc);
}

// ---------------- raw_A = Q K^T / sqrt(D) via V_WMMA_F32_16X16X4_F32 ----------------
__global__ void k_rawA(const float* __restrict__ qbuf, const float* __restrict__ kbuf,
                       float* __restrict__ Areg, float* __restrict__ sumsq) {
  int wave = (blockIdx.x * blockDim.x + threadIdx.x) >> 5;
  int lane = threadIdx.x & 31;
  int b = wave >> 12;            // 4096 16x16 tiles per batch
  int t = wave & 4095;
  if (b >= BB) return;
  int m0 = (t >> 6) << 4;
  int n0 = (t & 63) << 4;
  int lm = lane & 15;
  int lk = (lane >> 4) << 1;     // K offset 0 (lanes 0-15) or 2 (lanes 16-31)
  const float* qrow = qbuf + ((size_t)b * NB + m0 + lm) * DDIM;
  const float* krow = kbuf + ((size_t)b * NB + n0 + lm) * DDIM;
  v8f acc = {};
  #pragma unroll
  for (int k0 = 0; k0 < DDIM; k0 += 4) {
    v2f a, bf;
    a.x  = qrow[k0 + lk];     a.y  = qrow[k0 + lk + 1];
    bf.x = krow[k0 + lk];     bf.y = krow[k0 + lk + 1];
    acc = __builtin_amdgcn_wmma_f32_16x16x4_f32(false, a, false, bf,
                                                (short)0, acc, false, false);
  }
  const float scale = 0.17677669529663687f; // 1/sqrt(32)
  float ss = 0.f;
  #pragma unroll
  for (int r = 0; r < 8; ++r) {
    float v = acc[r] * scale;
    int row = m0 + r + ((lane >> 4) << 3);
    Areg[((size_t)b * NB + row) * NB + n0 + lm] = v;
    ss += v * v;
  }
  ss = wave_sum(ss);
  if (lane == 0) atomicAdd(&sumsq[b], ss);
}

// ------------- rms-norm raw_A per batch, EMA with A_ema, mask rows<2, clip+eps -------------
__global__ void k_ema(const float* __restrict__ A_ema, const float* __restrict__ sumsq,
                      const int* __restrict__ step, float* __restrict__ Areg) {
  float sc = (float)step[0] + 1.0f;
  const float invNN = 1.0f / ((float)NB * (float)NB);
  for (int idx = blockIdx.x * blockDim.x + threadIdx.x; idx < BNN;
       idx += gridDim.x * blockDim.x) {
    int b = idx / (NB * NB);
    int n = (idx / NB) & (NB - 1);
    float rawn = Areg[idx] * rsqrtf(sumsq[b] * invNN + 1e-8f);
    float val = (sc == 1.0f) ? rawn : (A_ema[idx] * 0.9f + rawn * 0.1f);
    if (n < 2) val = 0.f;
    Areg[idx] = fmaxf(val, 0.f) + 1e-8f;   // sinkhorn init: clip + eps
  }
}

// ---------------- sinkhorn: row normalize (wave per row) ----------------
__global__ void k_rownorm(float* __restrict__ A) {
  int wave = (blockIdx.x * blockDim.x + threadIdx.x) >> 5;
  int lane = threadIdx.x & 31;
  if (wave >= BB * NB) return;
  float* row = A + (size_t)wave * NB;
  float s = 0.f;
  #pragma unroll
  for (int t = 0; t < NB / 32; ++t) s += row[lane + t * 32];
  s = wave_sum(s);
  float inv = 1.0f / s;
  #pragma unroll
  for (int t = 0; t < NB / 32; ++t) row[lane + t * 32] *= inv;
}

// ---------------- sinkhorn: column partial sums ----------------
__global__ void k_colsum(const float* __restrict__ A, float* __restrict__ colsum) {
  int b = blockIdx.y;
  int chunk = blockIdx.x;        // 64 chunks of 16 rows
  int t = threadIdx.x;
  float loc[4] = {0.f, 0.f, 0.f, 0.f};
  for (int r = 0; r < 16; ++r) {
    const float* p = A + ((size_t)b * NB + chunk * 16 + r) * NB;
    #pragma unroll
    for (int j = 0; j < 4; ++j) loc[j] += p[t + j * 256];
  }
  #pragma unroll
  for (int j = 0; j < 4; ++j) atomicAdd(&colsum[b * NB + t + j * 256], loc[j]);
}

// ---------------- sinkhorn: column divide ----------------
__global__ void k_coldiv(float* __restrict__ A, const float* __restrict__ colsum) {
  for (int idx = blockIdx.x * blockDim.x + threadIdx.x; idx < BNN;
       idx += gridDim.x * blockDim.x) {
    int b = idx / (NB * NB);
    int col = idx & (NB - 1);
    A[idx] /= colsum[b * NB + col];
  }
}

// ---------------- threshold to sparse + eps (init of second sinkhorn) ----------------
__global__ void k_sparse(float* __restrict__ A) {
  const float thr = 0.5f / (float)NB;
  for (int idx = blockIdx.x * blockDim.x + threadIdx.x; idx < BNN;
       idx += gridDim.x * blockDim.x) {
    float v = A[idx];
    A[idx] = ((v > thr) ? v : 0.f) + 1e-8f;
  }
}

// ---------------- apply mask (rows<2), accumulate per-batch sum & sumsq for var ----------------
__global__ void k_maskvar(float* __restrict__ A, float* __restrict__ vsum,
                          float* __restrict__ vsumsq) {
  int b = blockIdx.y;
  float* Ab = A + (size_t)b * NB * NB;
  float s = 0.f, ss = 0.f;
  for (int i = blockIdx.x * blockDim.x + threadIdx.x; i < NB * NB;
       i += gridDim.x * blockDim.x) {
    float v = Ab[i];
    if (i < 2 * NB) { v = 0.f; Ab[i] = 0.f; }
    s += v; ss += v * v;
  }
  s = wave_sum(s); ss = wave_sum(ss);
  if ((threadIdx.x & 31) == 0) {
    atomicAdd(&vsum[b], s);
    atomicAdd(&vsumsq[b], ss);
  }
}

// ---------------- V = output + relu(0.01 - var)*noise_out ----------------
__global__ void k_makeV(const float* __restrict__ outp, const float* __restrict__ noise_out,
                        const float* __restrict__ vsum, const float* __restrict__ vsumsq,
                        float* __restrict__ V) {
  int idx = blockIdx.x * blockDim.x + threadIdx.x;
  if (idx >= BND) return;
  int b = idx / (NB * DDIM);
  const float invNN = 1.0f / ((float)NB * (float)NB);
  float mean = vsum[b] * invNN;
  float var = vsumsq[b] * invNN - mean * mean;
  float vd = fmaxf(0.01f - var, 0.f);
  V[idx] = outp[idx] + vd * noise_out[idx];
}

// ---------------- target = A @ V via V_WMMA_F32_16X16X4_F32 (K = 1024) ----------------
__global__ void k_target(const float* __restrict__ A, const float* __restrict__ V,
                         float* __restrict__ target) {
  int wave = (blockIdx.x * blockDim.x + threadIdx.x) >> 5;
  int lane = threadIdx.x & 31;
  if (wave >= BB * 64 * 2) return;   // 8 batches x 64 m-tiles x 2 n-tiles
  int b = wave >> 7;
  int mt = (wave >> 1) & 63;
  int nt = wave & 1;
  int m0 = mt << 4, n0 = nt << 4;
  int lm = lane & 15;
  int lk = (lane >> 4) << 1;
  const float* Arow = A + ((size_t)b * NB + m0 + lm) * NB;
  const float* Vb = V + (size_t)b * NB * DDIM;
  v8f acc = {};
  for (int kk = 0; kk < NB; kk += 4) {
    v2f a, bf;
    a.x  = Arow[kk + lk];
    a.y  = Arow[kk + lk + 1];
    bf.x = Vb[(kk + lk) * DDIM + n0 + lm];
    bf.y = Vb[(kk + lk + 1) * DDIM + n0 + lm];
    acc = __builtin_amdgcn_wmma_f32_16x16x4_f32(false, a, false, bf,
                                                (short)0, acc, false, false);
  }
  #pragma unroll
  for (int r = 0; r < 8; ++r) {
    int row = m0 + r + ((lane >> 4) << 3);
    target[((size_t)b * NB + row) * DDIM + n0 + lm] = acc[r];
  }
}

// ---------------- target rows 0,1 <- eye_env / stomach_env ----------------
__global__ void k_fixrows(const float* __restrict__ eye, const float* __restrict__ stomach,
                          float* __restrict__ target) {
  int idx = blockIdx.x * blockDim.x + threadIdx.x;
  if (idx >= BB * 2 * DDIM) return;
  int b = idx / (2 * DDIM);
  int r = (idx / DDIM) & 1;
  int d = idx & (DDIM - 1);
  target[((size_t)b * NB + r) * DDIM + d] = (r == 0) ? eye[b * DDIM + d]
                                                     : stomach[b * DDIM + d];
}

// ---------- fused err -> get_gradient -> step (wave per (weight, node)) ----------
__global__ void k_update(const float* __restrict__ state,
                         const float* __restrict__ pred, const float* __restrict__ target,
                         const float* __restrict__ kbuf, const float* __restrict__ qbuf,
                         const float* __restrict__ E1, const float* __restrict__ E2,
                         const float* __restrict__ E3,
                         const float* __restrict__ n1, const float* __restrict__ n2,
                         const float* __restrict__ n3,
                         const float* __restrict__ w1, const float* __restrict__ w2,
                         const float* __restrict__ w3,
                         const float* __restrict__ g1, const float* __restrict__ g2,
                         const float* __restrict__ g3,
                         float* __restrict__ w1n, float* __restrict__ w2n,
                         float* __restrict__ w3n,
                         const int* __restrict__ step) {
  int wave = (blockIdx.x * blockDim.x + threadIdx.x) >> 5;
  int lane = threadIdx.x & 31;
  int l = wave / (BB * NB);
  int node = wave - l * (BB * NB);
  if (l >= 3) return;
  int off = node * DDIM + lane;

  float pv = pred[off], tv = target[off], kv = kbuf[off], qv = qbuf[off];
  float e1b = pv - tv;
  float e2  = kv - e1b;
  float e3  = qv - e2;
  float e1  = e1b - e3;      // pred - (target + err3)
  float err = (l == 0) ? e1 : (l == 1) ? e2 : e3;

  const float* E  = (l == 0) ? E1 : (l == 1) ? E2 : E3;
  const float* nz = (l == 0) ? n1 : (l == 1) ? n2 : n3;
  const float* w  = (l == 0) ? w1 : (l == 1) ? w2 : w3;
  const float* g  = (l == 0) ? g1 : (l == 1) ? g2 : g3;
  float* wo       = (l == 0) ? w1n : (l == 1) ? w2n : w3n;

  // softmax over D (wave32 == D)
  float mx = wave_max(err);
  float ex = expf(err - mx);
  float s  = wave_sum(ex);
  float Ec = ex / s;
  float En = E[off] * 0.9f + 0.1f * Ec;       // EMA_SPEED = 0.1
  float adv = Ec - En;
  float plast = 1.0f + adv * rsqrtf(adv * adv + 1e-8f);  // 1 + rms_norm(single elem)
  float errsc = sqrtf(wave_sum(err * err) * (1.0f / DDIM) + 1e-8f);
  float noisy = state[off] + 0.01f * nz[off] * errsc;     // NOISE_SCALE = 0.01
  float msj = wave_sum(noisy * noisy) * (1.0f / DDIM);

  // wrapping gaussian kernel over last dim (j = lane)
  float sc = (float)step[0] + 1.0f;
  float center = fmodf(sc, (float)DDIM);                  // SPEED = 1.0
  float dist = fabsf((float)lane - center);
  dist = fminf(dist, (float)DDIM - dist);
  float kern = expf(-0.5f * (dist * 0.25f) * (dist * 0.25f)); // WIDTH = 4

  const float* wp = w + (size_t)node * DDIM * DDIM;
  const float* gp = g + (size_t)node * DDIM * DDIM;
  float* wop = wo + (size_t)node * DDIM * DDIM;

  float wn[DDIM];
  float ss2 = 0.f;
  #pragma unroll
  for (int i = 0; i < DDIM; ++i) {
    float ei = __shfl(err, i, 32);
    float pi = __shfl(plast, i, 32);
    float inv = rsqrtf(ei * ei * msj + 1e-8f);            // rms over j of err_i*noisy_j
    float grad = pi * (-(ei * noisy) * inv);
    float wv = wp[i * DDIM + lane];
    float gv = gp[i * DDIM + lane];
    grad = (grad - 0.01f * wv) * kern;                    // WD = 0.01, kernel mask
    float gn = 0.9f * gv + 0.1f * grad;                   // MOMENTUM = 0.9
    float wv2 = wv + 0.01f * gn;                          // LR = 0.01
    wn[i] = wv2;
    ss2 += wv2 * wv2;
  }
  ss2 = wave_sum(ss2);
  float nsc = rsqrtf(ss2 * (1.0f / (DDIM * DDIM)) + 1e-8f);  // rms_norm over (-1,-2), W_SCALE=1
  #pragma unroll
  for (int i = 0; i < DDIM; ++i)
    wop[i * DDIM + lane] = wn[i] * nsc;
}

extern "C" void kernel_launch(void* const* d_in, const int* in_sizes, int n_in,
                              void* d_out, int out_size, void* d_ws, size_t ws_size,
                              hipStream_t stream) {
  (void)in_sizes; (void)n_in; (void)out_size; (void)ws_size;
  const float* state     = (const float*)d_in[0];
  const float* outp      = (const float*)d_in[1];
  const float* A_ema     = (const float*)d_in[2];
  const float* w1        = (const float*)d_in[3];
  const float* w2        = (const float*)d_in[4];
  const float* w3        = (const float*)d_in[5];
  const float* E1        = (const float*)d_in[6];
  const float* E2        = (const float*)d_in[7];
  const float* E3        = (const float*)d_in[8];
  const float* g1        = (const float*)d_in[9];
  const float* g2        = (const float*)d_in[10];
  const float* g3        = (const float*)d_in[11];
  const float* eye       = (const float*)d_in[12];
  const float* stomach   = (const float*)d_in[13];
  const float* noise_out = (const float*)d_in[14];
  const float* n1        = (const float*)d_in[15];
  const float* n2        = (const float*)d_in[16];
  const float* n3        = (const float*)d_in[17];
  const int*   step      = (const int*)d_in[18];

  float* out    = (float*)d_out;
  float* target = out;                 // (B,N,D)
  float* pred   = out + BND;           // (B,N,D)
  float* Areg   = out + 2 * BND;       // (B,N,N)  -- also sinkhorn working buffer
  float* w1n    = Areg + BNN;
  float* w2n    = w1n + BNN;
  float* w3n    = w2n + BNN;

  float* ws     = (float*)d_ws;
  float* qbuf   = ws;                  // (B,N,D)
  float* kbuf   = ws + BND;            // (B,N,D)
  float* Vbuf   = ws + 2 * BND;        // (B,N,D)
  float* colsum = ws + 3 * BND;        // (B,N)
  float* sumsq  = colsum + BB * NB;    // 8 + 8 + 8 scalars
  float* vsum   = sumsq + BB;
  float* vsumsq = vsum + BB;

  // 1. q, k, pred
  k_piece<<<3072, 256, 0, stream>>>(state, w1, w2, w3, pred, kbuf, qbuf);
  // 2. zero scalar accumulators
  hipMemsetAsync(sumsq, 0, 3 * BB * sizeof(float), stream);
  // 3. raw_A (WMMA f32) + per-batch sumsq
  k_rawA<<<4096, 256, 0, stream>>>(qbuf, kbuf, Areg, sumsq);
  // 4. rms-norm + EMA + mask + clip/eps
  k_ema<<<4096, 256, 0, stream>>>(A_ema, sumsq, step, Areg);
  // 5-7. sinkhorn #1, sparsify, sinkhorn #2 (in place in Areg)
  for (int pass = 0; pass < 2; ++pass) {
    for (int it = 0; it < 10; ++it) {
      k_rownorm<<<1024, 256, 0, stream>>>(Areg);
      hipMemsetAsync(colsum, 0, BB * NB * sizeof(float), stream);
      k_colsum<<<dim3(64, BB), 256, 0, stream>>>(Areg, colsum);
      k_coldiv<<<4096, 256, 0, stream>>>(Areg, colsum);
    }
    if (pass == 0) k_sparse<<<4096, 256, 0, stream>>>(Areg);
  }
  // 8. final mask + variance accumulation
  k_maskvar<<<dim3(512, BB), 256, 0, stream>>>(Areg, vsum, vsumsq);
  // 9. V = output + vd*noise
  k_makeV<<<BND / 256, 256, 0, stream>>>(outp, noise_out, vsum, vsumsq, Vbuf);
  // 10. target = A @ V (WMMA f32)
  k_target<<<128, 256, 0, stream>>>(Areg, Vbuf, target);
  // 11. env rows
  k_fixrows<<<2, 256, 0, stream>>>(eye, stomach, target);
  // 12. fused gradient + weight update
  k_update<<<3072, 256, 0, stream>>>(state, pred, target, kbuf, qbuf,
                                     E1, E2, E3, n1, n2, n3,
                                     w1, w2, w3, g1, g2, g3,
                                     w1n, w2n, w3n, step);
}